// GCN_82497731822011
// MI455X (gfx1250) — compile-verified
//
#include <hip/hip_runtime.h>
#include <hip/hip_bf16.h>

typedef __attribute__((ext_vector_type(2))) float v2f;
typedef __attribute__((ext_vector_type(8))) float v8f;

#define IN_DIM  128
#define HID_DIM 128
#define OUT_DIM 3
#define LDS_STRIDE 132   // 128 + 4 pad: lane m -> bank 4m, conflict-free ds_load_b64

// ---------------- utility kernels ----------------

__global__ void k_zero_f4(float4* __restrict__ p, int n4) {
    int i = blockIdx.x * blockDim.x + threadIdx.x;
    if (i < n4) p[i] = make_float4(0.f, 0.f, 0.f, 0.f);
}

__global__ void k_deg_init(int* __restrict__ deg, int N) {
    int i = blockIdx.x * blockDim.x + threadIdx.x;
    if (i < N) deg[i] = 1;                       // self-loop
}

__global__ void k_deg_count(const int* __restrict__ dst, int* __restrict__ deg, int E) {
    int e = blockIdx.x * blockDim.x + threadIdx.x;
    if (e < E)
        __hip_atomic_fetch_add(&deg[dst[e]], 1, __ATOMIC_RELAXED, __HIP_MEMORY_SCOPE_AGENT);
}

__global__ void k_dinv(const int* __restrict__ deg, float* __restrict__ dinv, int N) {
    int i = blockIdx.x * blockDim.x + threadIdx.x;
    if (i < N) dinv[i] = rsqrtf((float)(deg[i] < 1 ? 1 : deg[i]));
}

// ---------------- GEMM1: XW = X @ W1 via V_WMMA_F32_16X16X4_F32 ----------------
// Block = 256 threads = 8 waves; block b covers rows [16b,16b+16), wave w cols [16w,16w+16).
// A tile (16x128, 8KB) staged once per block in LDS (padded stride -> conflict-free).
__global__ void k_gemm1_wmma(const float* __restrict__ X, const float* __restrict__ W,
                             float* __restrict__ XW, int N) {
    __shared__ float As[16 * LDS_STRIDE];

    const int t = threadIdx.x;
    const int rowBase = blockIdx.x * 16;

    // cooperative stage: 512 float4 slots, 2 per thread
    if (rowBase + 16 <= N) {
#pragma unroll
        for (int i = 0; i < 2; ++i) {
            int slot = t + i * 256;
            int r = slot >> 5;               // row 0..15
            int c = (slot & 31) << 2;        // col 0,4,...,124
            const float4 v = *(const float4*)(X + (size_t)(rowBase + r) * IN_DIM + c);
            *(float4*)(&As[r * LDS_STRIDE + c]) = v;
        }
    } else {
#pragma unroll
        for (int i = 0; i < 2; ++i) {
            int slot = t + i * 256;
            int r = slot >> 5;
            int c = (slot & 31) << 2;
            int rr = rowBase + r; if (rr >= N) rr = N - 1;
            const float4 v = *(const float4*)(X + (size_t)rr * IN_DIM + c);
            *(float4*)(&As[r * LDS_STRIDE + c]) = v;
        }
    }
    __syncthreads();

    const int lane  = t & 31;
    const int wave  = t >> 5;                    // 0..7 -> column tile
    const int m16   = lane & 15;                 // A row / B col within tile
    const int khalf = (lane >> 4) << 1;          // 0 or 2 : K sub-pair this half-wave owns
    const int col   = wave * 16 + m16;
    const float* __restrict__ arow = &As[m16 * LDS_STRIDE + khalf];

    v8f acc = {};
#pragma unroll
    for (int k = 0; k < IN_DIM; k += 4) {
        v2f a = *(const v2f*)(arow + k);         // ds_load_b64, conflict-free
        v2f b;
        b.x = W[(size_t)(k + khalf)     * HID_DIM + col];
        b.y = W[(size_t)(k + khalf + 1) * HID_DIM + col];
        acc = __builtin_amdgcn_wmma_f32_16x16x4_f32(false, a, false, b,
                                                    (short)0, acc, false, false);
    }

    // D layout: VGPR r of lane l -> (M = r + 8*(l>>4), N = l&15) within tile
    const int srow_base = rowBase + ((lane >> 4) << 3);
#pragma unroll
    for (int r = 0; r < 8; ++r) {
        int srow = srow_base + r;
        if (srow < N) XW[(size_t)srow * HID_DIM + col] = acc[r];
    }
}

// ---------------- layer-1 edge scatter: H[dst] += XW[src] * norm ----------------
// One wave per edge; each lane moves a float4 (128 floats / 32 lanes).
__global__ void k_scatter1(const float* __restrict__ XW, const int* __restrict__ src,
                           const int* __restrict__ dst, const float* __restrict__ dinv,
                           float* __restrict__ H, int E) {
    int w    = (blockIdx.x * blockDim.x + threadIdx.x) >> 5;
    int lane = threadIdx.x & 31;
    if (w >= E) return;
    int s = src[w], d = dst[w];
    float nrm = dinv[s] * dinv[d];
    const float4 m = *(const float4*)(XW + (size_t)s * HID_DIM + lane * 4);
    float* hp = H + (size_t)d * HID_DIM + lane * 4;
    __hip_atomic_fetch_add(hp + 0, m.x * nrm, __ATOMIC_RELAXED, __HIP_MEMORY_SCOPE_AGENT);
    __hip_atomic_fetch_add(hp + 1, m.y * nrm, __ATOMIC_RELAXED, __HIP_MEMORY_SCOPE_AGENT);
    __hip_atomic_fetch_add(hp + 2, m.z * nrm, __ATOMIC_RELAXED, __HIP_MEMORY_SCOPE_AGENT);
    __hip_atomic_fetch_add(hp + 3, m.w * nrm, __ATOMIC_RELAXED, __HIP_MEMORY_SCOPE_AGENT);
}

// ---------------- fold self-loop + bias, ReLU (in place on H) ----------------
__global__ void k_relu_bias(float* __restrict__ H, const float* __restrict__ XW,
                            const float* __restrict__ dinv, const float* __restrict__ b1,
                            int N) {
    int t = blockIdx.x * blockDim.x + threadIdx.x;       // one float4 per thread
    if (t >= N * (HID_DIM / 4)) return;
    int node = t >> 5, q = t & 31;
    float di = dinv[node];
    float sl = di * di;
    const float4 x4 = *(const float4*)(XW + (size_t)node * HID_DIM + q * 4);
    const float4 b4 = *(const float4*)(b1 + q * 4);
    float4 h4 = *(float4*)(H + (size_t)node * HID_DIM + q * 4);
    h4.x = fmaxf(fmaf(x4.x, sl, h4.x) + b4.x, 0.f);
    h4.y = fmaxf(fmaf(x4.y, sl, h4.y) + b4.y, 0.f);
    h4.z = fmaxf(fmaf(x4.z, sl, h4.z) + b4.z, 0.f);
    h4.w = fmaxf(fmaf(x4.w, sl, h4.w) + b4.w, 0.f);
    *(float4*)(H + (size_t)node * HID_DIM + q * 4) = h4;
}

// ---------------- layer 2: XW2 = H @ W2 ; OUT = self-loop + bias ----------------
// One wave per node; lane handles a float4 of the row, then 5-step xor reduction.
__global__ void k_layer2(const float* __restrict__ H, const float* __restrict__ W2,
                         const float* __restrict__ b2, const float* __restrict__ dinv,
                         float* __restrict__ XW2, float* __restrict__ OUT, int N) {
    int node = (blockIdx.x * blockDim.x + threadIdx.x) >> 5;
    int lane = threadIdx.x & 31;
    if (node >= N) return;
    const float4 h4 = *(const float4*)(H + (size_t)node * HID_DIM + lane * 4);
    const int k = lane * 4;
    float s0, s1, s2;
    s0 = h4.x * W2[(k + 0) * OUT_DIM + 0] + h4.y * W2[(k + 1) * OUT_DIM + 0]
       + h4.z * W2[(k + 2) * OUT_DIM + 0] + h4.w * W2[(k + 3) * OUT_DIM + 0];
    s1 = h4.x * W2[(k + 0) * OUT_DIM + 1] + h4.y * W2[(k + 1) * OUT_DIM + 1]
       + h4.z * W2[(k + 2) * OUT_DIM + 1] + h4.w * W2[(k + 3) * OUT_DIM + 1];
    s2 = h4.x * W2[(k + 0) * OUT_DIM + 2] + h4.y * W2[(k + 1) * OUT_DIM + 2]
       + h4.z * W2[(k + 2) * OUT_DIM + 2] + h4.w * W2[(k + 3) * OUT_DIM + 2];
#pragma unroll
    for (int off = 16; off > 0; off >>= 1) {
        s0 += __shfl_xor(s0, off, 32);
        s1 += __shfl_xor(s1, off, 32);
        s2 += __shfl_xor(s2, off, 32);
    }
    if (lane == 0) {
        float di = dinv[node];
        float sl = di * di;
        XW2[(size_t)node * OUT_DIM + 0] = s0;
        XW2[(size_t)node * OUT_DIM + 1] = s1;
        XW2[(size_t)node * OUT_DIM + 2] = s2;
        OUT[(size_t)node * OUT_DIM + 0] = fmaf(s0, sl, b2[0]);
        OUT[(size_t)node * OUT_DIM + 1] = fmaf(s1, sl, b2[1]);
        OUT[(size_t)node * OUT_DIM + 2] = fmaf(s2, sl, b2[2]);
    }
}

// ---------------- layer-2 edge scatter: OUT[dst] += XW2[src] * norm ----------------
__global__ void k_scatter2(const float* __restrict__ XW2, const int* __restrict__ src,
                           const int* __restrict__ dst, const float* __restrict__ dinv,
                           float* __restrict__ OUT, int E) {
    int e = blockIdx.x * blockDim.x + threadIdx.x;
    if (e >= E) return;
    int s = src[e], d = dst[e];
    float nrm = dinv[s] * dinv[d];
    const float* xp = XW2 + (size_t)s * OUT_DIM;
    float* op = OUT + (size_t)d * OUT_DIM;
    __hip_atomic_fetch_add(op + 0, xp[0] * nrm, __ATOMIC_RELAXED, __HIP_MEMORY_SCOPE_AGENT);
    __hip_atomic_fetch_add(op + 1, xp[1] * nrm, __ATOMIC_RELAXED, __HIP_MEMORY_SCOPE_AGENT);
    __hip_atomic_fetch_add(op + 2, xp[2] * nrm, __ATOMIC_RELAXED, __HIP_MEMORY_SCOPE_AGENT);
}

// ---------------- host ----------------

static inline size_t align256(size_t x) { return (x + 255) & ~(size_t)255; }

extern "C" void kernel_launch(void* const* d_in, const int* in_sizes, int n_in,
                              void* d_out, int out_size, void* d_ws, size_t ws_size,
                              hipStream_t stream) {
    const float* X  = (const float*)d_in[0];
    const int*   EI = (const int*)d_in[1];
    const float* W1 = (const float*)d_in[3];
    const float* b1 = (const float*)d_in[4];
    const float* W2 = (const float*)d_in[5];
    const float* b2 = (const float*)d_in[6];
    float* OUT = (float*)d_out;

    const int N = in_sizes[0] / IN_DIM;
    const int E = in_sizes[1] / 2;
    const int* src = EI;
    const int* dst = EI + E;

    // workspace carve-up
    char* ws = (char*)d_ws;
    size_t off = 0;
    float* XW  = (float*)(ws + off); off = align256(off + (size_t)N * HID_DIM * 4);
    float* H   = (float*)(ws + off); off = align256(off + (size_t)N * HID_DIM * 4);
    float* XW2 = (float*)(ws + off); off = align256(off + (size_t)N * OUT_DIM * 4);
    int*   DEG = (int*)  (ws + off); off = align256(off + (size_t)N * 4);
    float* DNV = (float*)(ws + off); off = align256(off + (size_t)N * 4);
    (void)ws_size; (void)n_in; (void)out_size;

    const int T = 256;
    const int nNodeBlk  = (N + T - 1) / T;                     // per-node
    const int nEdgeBlk  = (E + T - 1) / T;                     // per-edge
    const int nH4       = N * (HID_DIM / 4);                   // float4 count of H
    const int nH4Blk    = (nH4 + T - 1) / T;
    const int nWaveNode = (N * 32 + T - 1) / T;                // wave per node
    const int nWaveEdge = (int)(((size_t)E * 32 + T - 1) / T); // wave per edge
    const int nRowTiles = (N + 15) / 16;

    // degree / normalization
    k_deg_init <<<nNodeBlk, T, 0, stream>>>(DEG, N);
    k_deg_count<<<nEdgeBlk, T, 0, stream>>>(dst, DEG, E);
    k_dinv     <<<nNodeBlk, T, 0, stream>>>(DEG, DNV, N);

    // layer 1
    k_zero_f4   <<<nH4Blk, T, 0, stream>>>((float4*)H, nH4);
    k_gemm1_wmma<<<nRowTiles, T, 0, stream>>>(X, W1, XW, N);
    k_scatter1  <<<nWaveEdge, T, 0, stream>>>(XW, src, dst, DNV, H, E);
    k_relu_bias <<<nH4Blk, T, 0, stream>>>(H, XW, DNV, b1, N);

    // layer 2
    k_layer2  <<<nWaveNode, T, 0, stream>>>(H, W2, b2, DNV, XW2, OUT, N);
    k_scatter2<<<nEdgeBlk, T, 0, stream>>>(XW2, src, dst, DNV, OUT, E);
}